// Loss_Labels_34213709479939
// MI455X (gfx1250) — compile-verified
//
#include <hip/hip_runtime.h>

// ---------------------------------------------------------------------------
// MI455X (gfx1250) notes:
//  * wave32 only; warpSize == 32.
//  * Memory-bound streaming reduction: ~12 B/elem * 8.4M elems ~= 100 MB,
//    floor ~4.3 us at 23.3 TB/s. Use b128 loads + deep occupancy.
//  * Wave-level reduction implemented with V_WMMA_F32_16X16X4_F32:
//    A = all-ones (16x4), B holds one partial per lane (other slot 0).
//    D[m][n] = colsum(B)[n] for all m, and sum_n colsum[n] == sum of lane
//    partials regardless of the exact VGPR<->matrix-element mapping.
// ---------------------------------------------------------------------------

typedef __attribute__((ext_vector_type(2))) float v2f;
typedef __attribute__((ext_vector_type(8))) float v8f;

__device__ __forceinline__ float stable_softplus(float d) {
    // log(1 + exp(d)) computed stably; this equals -log_softmax selected term.
    return fmaxf(d, 0.0f) + log1pf(expf(-fabsf(d)));
}

__device__ __forceinline__ float elem_loss(float s, float a, int lab) {
    // label==1 -> -log p(syn) = softplus(a - s)
    // label==2 -> -log p(ant) = softplus(s - a)
    // label==0 -> 0
    float d = (lab == 1) ? (a - s) : (s - a);
    float r = stable_softplus(d);
    return (lab == 1 || lab == 2) ? r : 0.0f;
}

// Sum x across all 32 lanes of the wave; result valid in every lane.
// Uses v_wmma_f32_16x16x4_f32 with an all-ones A matrix, then folds the
// 16 column-sums with 4 xor-shuffles.
__device__ __forceinline__ float wave_reduce32(float x) {
    v2f a; a[0] = 1.0f; a[1] = 1.0f;   // all 64 elements of A(16x4) == 1
    v2f b; b[0] = x;    b[1] = 0.0f;   // each lane's partial occupies 1 slot
    v8f c = {};
    // 8 args: (neg_a, A, neg_b, B, c_mod, C, reuse_a, reuse_b)
    c = __builtin_amdgcn_wmma_f32_16x16x4_f32(
        false, a, false, b, (short)0, c, false, false);
    float s = c[0];                    // colsum[N], N = lane (mod 16)
    s += __shfl_xor(s, 8, 32);
    s += __shfl_xor(s, 4, 32);
    s += __shfl_xor(s, 2, 32);
    s += __shfl_xor(s, 1, 32);
    return s;                          // = sum of all 32 lane partials
}

// Deterministic block reduction: WMMA wave reduce -> LDS -> thread 0.
__device__ __forceinline__ float block_reduce(float acc) {
    __shared__ float smem[32];
    float wsum = wave_reduce32(acc);
    const int wave = threadIdx.x >> 5;
    const int lane = threadIdx.x & 31;
    if (lane == 0) smem[wave] = wsum;
    __syncthreads();
    float t = 0.0f;
    if (threadIdx.x == 0) {
        const int nwaves = blockDim.x >> 5;
        for (int w = 0; w < nwaves; ++w) t += smem[w];  // fixed order
    }
    return t;  // valid in thread 0 only
}

__global__ void loss_partial_kernel(const float* __restrict__ syn,
                                    const float* __restrict__ ant,
                                    const int* __restrict__ lab,
                                    float* __restrict__ partials,
                                    int n) {
    const int tid    = blockIdx.x * blockDim.x + threadIdx.x;
    const int stride = gridDim.x * blockDim.x;

    float acc = 0.0f;

    const int nvec = n >> 2;  // float4 / int4 chunks
    const float4* __restrict__ s4 = (const float4*)syn;
    const float4* __restrict__ a4 = (const float4*)ant;
    const int4*   __restrict__ l4 = (const int4*)lab;

    for (int v = tid; v < nvec; v += stride) {
        float4 s = s4[v];
        float4 a = a4[v];
        int4   l = l4[v];
        acc += elem_loss(s.x, a.x, l.x);
        acc += elem_loss(s.y, a.y, l.y);
        acc += elem_loss(s.z, a.z, l.z);
        acc += elem_loss(s.w, a.w, l.w);
    }
    // scalar tail (B is divisible by 4, but stay generic)
    for (int i = (nvec << 2) + tid; i < n; i += stride)
        acc += elem_loss(syn[i], ant[i], lab[i]);

    // All threads reach here (no divergent exits) -> EXEC all-1s for WMMA.
    float t = block_reduce(acc);
    if (threadIdx.x == 0) partials[blockIdx.x] = t;
}

__global__ void loss_final_kernel(const float* __restrict__ partials,
                                  int nparts,
                                  float* __restrict__ out,
                                  float inv_n) {
    float acc = 0.0f;
    for (int i = threadIdx.x; i < nparts; i += blockDim.x)
        acc += partials[i];
    float t = block_reduce(acc);
    if (threadIdx.x == 0) out[0] = t * inv_n;
}

extern "C" void kernel_launch(void* const* d_in, const int* in_sizes, int n_in,
                              void* d_out, int out_size, void* d_ws, size_t ws_size,
                              hipStream_t stream) {
    const float* syn = (const float*)d_in[0];
    const float* ant = (const float*)d_in[1];
    const int*   lab = (const int*)d_in[2];
    float* out = (float*)d_out;
    const int n = in_sizes[0];

    const int BLOCK = 256;   // 8 waves per workgroup (wave32)
    const int GRID  = 2048;  // 524288 threads -> 4 float4 chunks per thread

    float* partials = (float*)d_ws;  // 2048 floats of scratch

    loss_partial_kernel<<<GRID, BLOCK, 0, stream>>>(syn, ant, lab, partials, n);
    loss_final_kernel<<<1, BLOCK, 0, stream>>>(partials, GRID, out,
                                               1.0f / (float)n);
}